// GINConv_21912923144580
// MI455X (gfx1250) — compile-verified
//
#include <hip/hip_runtime.h>

typedef __attribute__((ext_vector_type(2))) float v2f;
typedef __attribute__((ext_vector_type(8))) float v8f;

#define D 128
#define ROWS 64          // rows of x handled per workgroup (4 row-tiles of 16)
#define APAD 132         // padded LDS row stride (bank-conflict-free fragments)
#define THREADS 256

// ---------------------------------------------------------------------------
// Kernel 1: zero the aggregation buffer (atomics accumulate, so every call
// must start from zero; harness does not re-poison d_ws between replays).
// ---------------------------------------------------------------------------
__global__ void gin_zero_kernel(float* __restrict__ p, int n4) {
    int i = blockIdx.x * blockDim.x + threadIdx.x;
    if (i < n4) ((float4*)p)[i] = make_float4(0.f, 0.f, 0.f, 0.f);
}

// ---------------------------------------------------------------------------
// Kernel 2: edge scatter-add.  One wave32 per edge; each lane moves 4 floats
// (float4 gather of x[src], 4x global_atomic_add_f32 into agg[dst]).
// edge_index is int64 in the reference -> long long.
// ---------------------------------------------------------------------------
__global__ void gin_scatter_kernel(const float* __restrict__ x,
                                   const long long* __restrict__ ei,
                                   float* __restrict__ agg, int E) {
    int warp = (blockIdx.x * blockDim.x + threadIdx.x) >> 5;
    int lane = threadIdx.x & 31;
    if (warp >= E) return;
    int src = (int)ei[warp];
    int dst = (int)ei[(size_t)E + warp];
    float4 v = ((const float4*)(x + (size_t)src * D))[lane];
    float* a = agg + (size_t)dst * D + lane * 4;
    atomicAdd(a + 0, v.x);
    atomicAdd(a + 1, v.y);
    atomicAdd(a + 2, v.z);
    atomicAdd(a + 3, v.w);
}

// ---------------------------------------------------------------------------
// Kernel 3: fused MLP.  Per 64-row block:
//   sA  = (1+eps)*x + agg              (LDS, padded stride)
//   sH  = relu(sA @ W1 + b1)           (GEMM1, v_wmma_f32_16x16x4_f32)
//   out = sH @ W2 + b2                 (GEMM2, v_wmma_f32_16x16x4_f32)
// Weights staged TRANSPOSED in LDS so each B fragment is one ds_load_b64.
// k-outer / row-tile-inner with 4 live accumulators reuses each B fragment
// across all 4 row-tiles (1 B + 4 A loads per 4 WMMAs).
// ---------------------------------------------------------------------------
__global__ __launch_bounds__(THREADS) void gin_mlp_kernel(
    const float* __restrict__ x, const float* __restrict__ agg,
    const float* __restrict__ W1, const float* __restrict__ b1,
    const float* __restrict__ W2, const float* __restrict__ b2,
    const float* __restrict__ eps, float* __restrict__ out, int N) {

    __shared__ float sW1t[D * APAD];      // W1^T, padded: 66 KB
    __shared__ float sW2t[D * APAD];      // W2^T, padded: 66 KB
    __shared__ float sA[ROWS * APAD];     // 33 KB
    __shared__ float sH[ROWS * APAD];     // 33 KB

    const int tid  = threadIdx.x;
    const int lane = tid & 31;
    const int w    = tid >> 5;
    const int r0   = blockIdx.x * ROWS;
    const float eps1 = 1.0f + eps[0];

    // Stage weights transposed: sWt[col*APAD + k] = W[k*D + col].
    for (int i = tid; i < D * (D / 4); i += THREADS) {
        int k = i >> 5;            // D/4 == 32 float4 chunks per row
        int c = (i & 31) * 4;
        float4 w1 = ((const float4*)W1)[i];
        float4 w2 = ((const float4*)W2)[i];
        sW1t[(c + 0) * APAD + k] = w1.x;
        sW1t[(c + 1) * APAD + k] = w1.y;
        sW1t[(c + 2) * APAD + k] = w1.z;
        sW1t[(c + 3) * APAD + k] = w1.w;
        sW2t[(c + 0) * APAD + k] = w2.x;
        sW2t[(c + 1) * APAD + k] = w2.y;
        sW2t[(c + 2) * APAD + k] = w2.z;
        sW2t[(c + 3) * APAD + k] = w2.w;
    }
    // Stage A = (1+eps)*x + agg, zero-padded past N.
    for (int i = tid; i < ROWS * (D / 4); i += THREADS) {
        int r = i >> 5;
        int c = (i & 31) * 4;
        int row = r0 + r;
        float4 v = make_float4(0.f, 0.f, 0.f, 0.f);
        if (row < N) {
            float4 xv = *(const float4*)(x   + (size_t)row * D + c);
            float4 av = *(const float4*)(agg + (size_t)row * D + c);
            v.x = eps1 * xv.x + av.x;
            v.y = eps1 * xv.y + av.y;
            v.z = eps1 * xv.z + av.z;
            v.w = eps1 * xv.w + av.w;
        }
        *(float4*)(sA + r * APAD + c) = v;
    }
    __syncthreads();

    // Fragment coordinates per ISA 7.12.2 (fp32, wave32):
    //   A 16x4 : lane l -> M = l&15, holds K = k0+2*(l>>4), +1
    //   B 4x16 : lane l -> N = l&15, holds same K pair
    //   C/D    : VGPR r, lane l -> M = r + 8*(l>>4), N = l&15
    const int m     = lane & 15;
    const int kk    = (lane >> 4) * 2;
    const int col   = w * 16 + (lane & 15);
    const int rhalf = (lane >> 4) * 8;

    // ---- GEMM1 + bias + ReLU -> sH ----
    {
        v8f acc[4] = {};
        const float* Bp = sW1t + col * APAD + kk;
        const float* A0 = sA + (0 * 16 + m) * APAD + kk;
        const float* A1 = sA + (1 * 16 + m) * APAD + kk;
        const float* A2 = sA + (2 * 16 + m) * APAD + kk;
        const float* A3 = sA + (3 * 16 + m) * APAD + kk;
#pragma unroll
        for (int k0 = 0; k0 < D; k0 += 4) {
            v2f b = *(const v2f*)(Bp + k0);   // one ds_load_b64, reused 4x
            acc[0] = __builtin_amdgcn_wmma_f32_16x16x4_f32(
                false, *(const v2f*)(A0 + k0), false, b, (short)0, acc[0], false, false);
            acc[1] = __builtin_amdgcn_wmma_f32_16x16x4_f32(
                false, *(const v2f*)(A1 + k0), false, b, (short)0, acc[1], false, false);
            acc[2] = __builtin_amdgcn_wmma_f32_16x16x4_f32(
                false, *(const v2f*)(A2 + k0), false, b, (short)0, acc[2], false, false);
            acc[3] = __builtin_amdgcn_wmma_f32_16x16x4_f32(
                false, *(const v2f*)(A3 + k0), false, b, (short)0, acc[3], false, false);
        }
        const float bias1 = b1[col];
#pragma unroll
        for (int rt = 0; rt < 4; ++rt) {
#pragma unroll
            for (int r = 0; r < 8; ++r) {
                float v = acc[rt][r] + bias1;
                sH[(rt * 16 + rhalf + r) * APAD + col] = (v > 0.f) ? v : 0.f;
            }
        }
    }
    __syncthreads();

    // ---- GEMM2 + bias -> global out ----
    {
        v8f acc[4] = {};
        const float* Bp = sW2t + col * APAD + kk;
        const float* H0 = sH + (0 * 16 + m) * APAD + kk;
        const float* H1 = sH + (1 * 16 + m) * APAD + kk;
        const float* H2 = sH + (2 * 16 + m) * APAD + kk;
        const float* H3 = sH + (3 * 16 + m) * APAD + kk;
#pragma unroll
        for (int k0 = 0; k0 < D; k0 += 4) {
            v2f b = *(const v2f*)(Bp + k0);
            acc[0] = __builtin_amdgcn_wmma_f32_16x16x4_f32(
                false, *(const v2f*)(H0 + k0), false, b, (short)0, acc[0], false, false);
            acc[1] = __builtin_amdgcn_wmma_f32_16x16x4_f32(
                false, *(const v2f*)(H1 + k0), false, b, (short)0, acc[1], false, false);
            acc[2] = __builtin_amdgcn_wmma_f32_16x16x4_f32(
                false, *(const v2f*)(H2 + k0), false, b, (short)0, acc[2], false, false);
            acc[3] = __builtin_amdgcn_wmma_f32_16x16x4_f32(
                false, *(const v2f*)(H3 + k0), false, b, (short)0, acc[3], false, false);
        }
        const float bias2 = b2[col];
#pragma unroll
        for (int rt = 0; rt < 4; ++rt) {
#pragma unroll
            for (int r = 0; r < 8; ++r) {
                int row = r0 + rt * 16 + rhalf + r;
                if (row < N) out[(size_t)row * D + col] = acc[rt][r] + bias2;
            }
        }
    }
}

// ---------------------------------------------------------------------------
extern "C" void kernel_launch(void* const* d_in, const int* in_sizes, int n_in,
                              void* d_out, int out_size, void* d_ws, size_t ws_size,
                              hipStream_t stream) {
    const float*     x   = (const float*)d_in[0];
    const long long* ei  = (const long long*)d_in[1];
    const float*     W1  = (const float*)d_in[2];
    const float*     b1  = (const float*)d_in[3];
    const float*     W2  = (const float*)d_in[4];
    const float*     b2  = (const float*)d_in[5];
    const float*     eps = (const float*)d_in[6];
    float*           out = (float*)d_out;

    const int N = in_sizes[0] / D;     // 100000
    const int E = in_sizes[1] / 2;     // 1600000
    float* agg = (float*)d_ws;         // N*D floats of scratch

    // 1) agg = 0
    int n4 = (N * D) / 4;
    gin_zero_kernel<<<(n4 + 255) / 256, 256, 0, stream>>>(agg, n4);

    // 2) agg[dst] += x[src]   (one wave per edge, 8 waves per block)
    int sblocks = (E + 7) / 8;
    gin_scatter_kernel<<<sblocks, 256, 0, stream>>>(x, ei, agg, E);

    // 3) fused eps-residual + MLP via WMMA
    int gblocks = (N + ROWS - 1) / ROWS;
    gin_mlp_kernel<<<gblocks, THREADS, 0, stream>>>(x, agg, W1, b1, W2, b2,
                                                    eps, out, N);
}